// PositionLinearAttention_14431090115169
// MI455X (gfx1250) — compile-verified
//
#include <hip/hip_runtime.h>
#include <hip/hip_bf16.h>

typedef float v2f __attribute__((ext_vector_type(2)));
typedef float v8f __attribute__((ext_vector_type(8)));

namespace {

constexpr int kB = 8;
constexpr int kC = 64;
constexpr int kM = 8;
constexpr int kN = 32768;            // 32*32*32 spatial positions
constexpr float kEps = 1e-6f;

constexpr int kChunks = 64;                          // blocks per batch
constexpr int kPosPerBlock = kN / kChunks;           // 512
constexpr int kWaves = 8;                            // 256 threads = 8 x wave32
constexpr int kTilesPerWave = kPosPerBlock / (kWaves * 16);  // 4

// ws layout (floats): kv_raw[B][8][64] | ksum[B][8] | vsum_raw[B][64]
constexpr int kKvOff = 0;
constexpr int kKsOff = kB * kM * kC;         // 4096
constexpr int kVsOff = kKsOff + kB * kM;     // 4160
constexpr int kWsFloats = kVsOff + kB * kC;  // 4672

// padded strides (bank-conflict avoidance on 64 x 4B banks)
constexpr int kWS = 65;   // weight tables: row stride
constexpr int kTS = 17;   // K transpose staging: row stride

__device__ __forceinline__ v8f wmma_f32(v2f a, v2f b, v8f c) {
  // D = A(16x4) * B(4x16) + C, all f32 (CDNA5 V_WMMA_F32_16X16X4_F32)
  return __builtin_amdgcn_wmma_f32_16x16x4_f32(false, a, false, b, (short)0, c,
                                               false, false);
}

__global__ void zero_ws_kernel(float* __restrict__ ws) {
  int i = blockIdx.x * blockDim.x + threadIdx.x;
  if (i < kWsFloats) ws[i] = 0.0f;
}

// ---------------------------------------------------------------------------
// Pass 1: per-batch reductions with V-bias folded out:
//   kv_raw[m][c]  = sum_n K[m,n] * (Wv x)[c,n]
//   vsum_raw[c]   = sum_n (Wv x)[c,n]        (via ones-row in the kv WMMA)
//   ksum[m]       = sum_n K[m,n]
// K = l2norm(Wk x + bk). All contractions on the WMMA pipe.
// ---------------------------------------------------------------------------
__global__ __launch_bounds__(256)
__attribute__((amdgpu_waves_per_eu(4))) void pla_pass1(
    const float* __restrict__ x, const float* __restrict__ Wk,
    const float* __restrict__ bk, const float* __restrict__ Wv,
    float* __restrict__ ws) {
  __shared__ float sWv[kC * kWS];            // Wv, padded rows (~16.6 KB)
  __shared__ float sWk16[16 * kWS];          // Wk rows 0..7, rows 8..15 zero
  __shared__ float sVT[kWaves][16][65];      // per-wave V^T staging (n x c)
  __shared__ float sKT[kWaves][16][kTS];     // per-wave [K^T | 1 | 0] (n x m16)
  __shared__ float sKV[kM * kC];
  __shared__ float sKS[kM];
  __shared__ float sVS[kC];

  const int tid = threadIdx.x;
  const int wave = tid >> 5;
  const int lane = tid & 31;
  const int ln = lane & 15;
  const int half = lane >> 4;
  const int b = blockIdx.y;

  for (int i = tid; i < kC * kWS; i += 256) {
    const int r = i / kWS, col = i - r * kWS;
    sWv[i] = (col < kC) ? Wv[r * kC + col] : 0.0f;
  }
  for (int i = tid; i < 16 * kWS; i += 256) {
    const int r = i / kWS, col = i - r * kWS;
    sWk16[i] = (r < kM && col < kC) ? Wk[r * kC + col] : 0.0f;
  }
  // K-staging constant rows: m=8 -> 1.0 (vsum ones-row), m=9..15 -> 0.
  {
    // 16 positions x 8 rows = 128 entries per wave; 4 per lane.
#pragma unroll
    for (int j = 0; j < 4; ++j) {
      const int e = j * 32 + lane;
      const int p = e >> 3;
      const int m = 8 + (e & 7);
      sKT[wave][p][m] = (m == 8) ? 1.0f : 0.0f;
    }
  }
  for (int i = tid; i < kM * kC; i += 256) sKV[i] = 0.0f;
  if (tid < kM) sKS[tid] = 0.0f;
  if (tid < kC) sVS[tid] = 0.0f;
  __syncthreads();

  float bkr[8];
#pragma unroll
  for (int v = 0; v < 8; ++v) bkr[v] = bk[v];

  v8f kvacc[4];
#pragma unroll
  for (int cb = 0; cb < 4; ++cb) kvacc[cb] = (v8f){0, 0, 0, 0, 0, 0, 0, 0};
  float ksump[8] = {};

  const float* xb = x + (size_t)b * kC * kN;
  const int posBase = blockIdx.x * kPosPerBlock;

#pragma unroll 1
  for (int t = 0; t < kTilesPerWave; ++t) {
    const int n0 = posBase + (wave * kTilesPerWave + t) * 16;
    const int n = n0 + ln;

    // B-operands: 16-position tile of X, channels along the WMMA K dim.
    v2f breg[16];
#pragma unroll
    for (int ks = 0; ks < 16; ++ks) {
      const int kc = 4 * ks + 2 * half;
      breg[ks].x = xb[(size_t)kc * kN + n];
      breg[ks].y = xb[(size_t)(kc + 1) * kN + n];
    }

    // ---- K projection + L2 norm --------------------------------------
    v8f kacc = (v8f){0, 0, 0, 0, 0, 0, 0, 0};
#pragma unroll
    for (int ks = 0; ks < 16; ++ks) {
      const int kc = 4 * ks + 2 * half;
      v2f a;
      a.x = sWk16[ln * kWS + kc];
      a.y = sWk16[ln * kWS + kc + 1];
      kacc = wmma_f32(a, breg[ks], kacc);
    }
    float kn[8];
    float ss = 0.0f;
#pragma unroll
    for (int v = 0; v < 8; ++v) {
      kn[v] = kacc[v] + bkr[v];
      ss += kn[v] * kn[v];
    }
    const float rn = rsqrtf(ss);
#pragma unroll
    for (int v = 0; v < 8; ++v) kn[v] *= rn;
    if (half == 0) {  // rows m=0..7 live in lanes 0..15
#pragma unroll
      for (int v = 0; v < 8; ++v) {
        sKT[wave][ln][v] = kn[v];
        ksump[v] += kn[v];
      }
    }

    // ---- V_raw projection (4 c-blocks of 16), staged transposed ------
#pragma unroll
    for (int cb = 0; cb < 4; ++cb) {
      v8f acc = (v8f){0, 0, 0, 0, 0, 0, 0, 0};
#pragma unroll
      for (int ks = 0; ks < 16; ++ks) {
        const int kc = 4 * ks + 2 * half;
        const int row = cb * 16 + ln;
        v2f a;
        a.x = sWv[row * kWS + kc];
        a.y = sWv[row * kWS + kc + 1];
        acc = wmma_f32(a, breg[ks], acc);
      }
#pragma unroll
      for (int v = 0; v < 8; ++v)
        sVT[wave][ln][cb * 16 + v + 8 * half] = acc[v];
    }

    // ---- kv_raw += [K ; 1 ; 0] * V_raw^T over this tile ---------------
    // A rows 0..7 = K, row 8 = ones (yields vsum_raw in D row 8).
#pragma unroll
    for (int cb = 0; cb < 4; ++cb) {
#pragma unroll
      for (int ks = 0; ks < 4; ++ks) {
        const int p = 4 * ks + 2 * half;
        v2f a;
        a.x = sKT[wave][p][ln];
        a.y = sKT[wave][p + 1][ln];
        v2f bb;
        bb.x = sVT[wave][p][cb * 16 + ln];
        bb.y = sVT[wave][p + 1][cb * 16 + ln];
        kvacc[cb] = wmma_f32(a, bb, kvacc[cb]);
      }
    }
  }

  // ---- block-level reduction, then global atomics -----------------------
  if (half == 0) {  // D rows 0..7: kv_raw[m][c]
#pragma unroll
    for (int cb = 0; cb < 4; ++cb)
#pragma unroll
      for (int v = 0; v < 8; ++v)
        atomicAdd(&sKV[v * kC + cb * 16 + ln], kvacc[cb][v]);
#pragma unroll
    for (int v = 0; v < 8; ++v) atomicAdd(&sKS[v], ksump[v]);
  } else {  // D row 8 (v==0): vsum_raw[c]
#pragma unroll
    for (int cb = 0; cb < 4; ++cb)
      atomicAdd(&sVS[cb * 16 + ln], kvacc[cb][0]);
  }
  __syncthreads();

  float* kvg = ws + kKvOff + b * kM * kC;
  for (int i = tid; i < kM * kC; i += 256) atomicAdd(&kvg[i], sKV[i]);
  if (tid < kM) atomicAdd(&ws[kKsOff + b * kM + tid], sKS[tid]);
  if (tid < kC) atomicAdd(&ws[kVsOff + b * kC + tid], sVS[tid]);
}

// ---------------------------------------------------------------------------
// Pass 2: Q = l2norm(Wq x + bq); tailor = 1/(N + Q.(ksum+eps));
//         out = x + gamma * tailor * (vsum + kv^T Q)
// with kv = kv_raw + bv (x) ksum,  vsum = vsum_raw + N*bv  (bias fold-in).
// ---------------------------------------------------------------------------
__global__ __launch_bounds__(256)
__attribute__((amdgpu_waves_per_eu(4))) void pla_pass2(
    const float* __restrict__ x, const float* __restrict__ gamma,
    const float* __restrict__ Wq, const float* __restrict__ bq,
    const float* __restrict__ bv, const float* __restrict__ ws,
    float* __restrict__ out) {
  __shared__ float sWq16[16 * kWS];        // Wq rows 0..7, rows 8..15 zero
  __shared__ float sQT[kWaves][16][9];     // per-wave Q^T staging (n x m)
  __shared__ float sT[kWaves][16];

  const int tid = threadIdx.x;
  const int wave = tid >> 5;
  const int lane = tid & 31;
  const int ln = lane & 15;
  const int half = lane >> 4;
  const int b = blockIdx.y;

  for (int i = tid; i < 16 * kWS; i += 256) {
    const int r = i / kWS, col = i - r * kWS;
    sWq16[i] = (r < kM && col < kC) ? Wq[r * kC + col] : 0.0f;
  }
  __syncthreads();

  float ksums[8], ksr[8], bqr[8];
#pragma unroll
  for (int v = 0; v < 8; ++v) {
    ksums[v] = ws[kKsOff + b * kM + v];
    ksr[v] = ksums[v] + kEps;
    bqr[v] = bq[v];
  }
  const float* kvg = ws + kKvOff + b * kM * kC;
  v2f kva[4][2];  // A-operands of G = kv^T * Q (rows = c, K-dim = m)
#pragma unroll
  for (int cb = 0; cb < 4; ++cb)
#pragma unroll
    for (int ks = 0; ks < 2; ++ks) {
      const int m0 = 4 * ks + 2 * half;
      const int c = cb * 16 + ln;
      const float bvc = bv[c];
      kva[cb][ks].x = kvg[m0 * kC + c] + bvc * ksums[m0];
      kva[cb][ks].y = kvg[(m0 + 1) * kC + c] + bvc * ksums[m0 + 1];
    }
  float vsr[4][8];
#pragma unroll
  for (int cb = 0; cb < 4; ++cb)
#pragma unroll
    for (int v = 0; v < 8; ++v) {
      const int c = cb * 16 + v + 8 * half;
      vsr[cb][v] = ws[kVsOff + b * kC + c] + (float)kN * bv[c];
    }
  const float gam = gamma[0];

  const float* xb = x + (size_t)b * kC * kN;
  float* ob = out + (size_t)b * kC * kN;
  const int posBase = blockIdx.x * kPosPerBlock;

#pragma unroll 1
  for (int t = 0; t < kTilesPerWave; ++t) {
    const int n0 = posBase + (wave * kTilesPerWave + t) * 16;
    const int n = n0 + ln;

    v2f breg[16];
#pragma unroll
    for (int ks = 0; ks < 16; ++ks) {
      const int kc = 4 * ks + 2 * half;
      breg[ks].x = xb[(size_t)kc * kN + n];
      breg[ks].y = xb[(size_t)(kc + 1) * kN + n];
    }

    // ---- Q projection + L2 norm + tailor ----------------------------
    v8f qacc = (v8f){0, 0, 0, 0, 0, 0, 0, 0};
#pragma unroll
    for (int ks = 0; ks < 16; ++ks) {
      const int kc = 4 * ks + 2 * half;
      v2f a;
      a.x = sWq16[ln * kWS + kc];
      a.y = sWq16[ln * kWS + kc + 1];
      qacc = wmma_f32(a, breg[ks], qacc);
    }
    float qn[8];
    float ss = 0.0f;
#pragma unroll
    for (int v = 0; v < 8; ++v) {
      qn[v] = qacc[v] + bqr[v];
      ss += qn[v] * qn[v];
    }
    const float rn = rsqrtf(ss);
    float s = 0.0f;
#pragma unroll
    for (int v = 0; v < 8; ++v) {
      qn[v] *= rn;
      s += qn[v] * ksr[v];
    }
    const float tl = 1.0f / ((float)kN + s);
    if (half == 0) {
#pragma unroll
      for (int v = 0; v < 8; ++v) sQT[wave][ln][v] = qn[v];
      sT[wave][ln] = tl;
    }

    // ---- G = kv^T Q, fused output combine ---------------------------
    v2f qb[2];
#pragma unroll
    for (int ks = 0; ks < 2; ++ks) {
      const int m0 = 4 * ks + 2 * half;
      qb[ks].x = sQT[wave][ln][m0];
      qb[ks].y = sQT[wave][ln][m0 + 1];
    }
    const float tn = sT[wave][ln];
#pragma unroll
    for (int cb = 0; cb < 4; ++cb) {
      v8f g = (v8f){0, 0, 0, 0, 0, 0, 0, 0};
      g = wmma_f32(kva[cb][0], qb[0], g);
      g = wmma_f32(kva[cb][1], qb[1], g);
#pragma unroll
      for (int v = 0; v < 8; ++v) {
        const int c = cb * 16 + v + 8 * half;
        const size_t idx = (size_t)c * kN + n;
        ob[idx] = xb[idx] + gam * tn * (vsr[cb][v] + g[v]);
      }
    }
  }
}

}  // namespace

extern "C" void kernel_launch(void* const* d_in, const int* in_sizes, int n_in,
                              void* d_out, int out_size, void* d_ws,
                              size_t ws_size, hipStream_t stream) {
  (void)in_sizes; (void)n_in; (void)out_size; (void)ws_size;
  const float* x = (const float*)d_in[0];
  const float* gamma = (const float*)d_in[1];
  const float* Wq = (const float*)d_in[2];
  const float* bq = (const float*)d_in[3];
  const float* Wk = (const float*)d_in[4];
  const float* bk = (const float*)d_in[5];
  const float* Wv = (const float*)d_in[6];
  const float* bv = (const float*)d_in[7];
  float* out = (float*)d_out;
  float* ws = (float*)d_ws;

  zero_ws_kernel<<<(kWsFloats + 255) / 256, 256, 0, stream>>>(ws);
  dim3 grid(kChunks, kB);
  pla_pass1<<<grid, 256, 0, stream>>>(x, Wk, bk, Wv, ws);
  pla_pass2<<<grid, 256, 0, stream>>>(x, gamma, Wq, bq, bv, ws, out);
}